// Decoder_68152541053025
// MI455X (gfx1250) — compile-verified
//
#include <hip/hip_runtime.h>

// ---------------------------------------------------------------------------
// Latent-ODE decoder for MI455X (gfx1250, wave32).
//   out = [ predicted_x (B,T,128) | predicted_z (B,T,2) | a | b | c | d ]
// ---------------------------------------------------------------------------

typedef float v2f __attribute__((ext_vector_type(2)));
typedef float v8f __attribute__((ext_vector_type(8)));

constexpr int BATCH = 8192;
constexpr int TSTEP = 100;
constexpr int LAT   = 64;
constexpr int HID   = 200;
constexpr int INX   = 128;

__device__ __forceinline__ float softplus_f(float v) {
    // numerically stable softplus
    return fmaxf(v, 0.0f) + log1pf(__expf(-fabsf(v)));
}

// ---- CDNA5 async global->LDS copy (ASYNCcnt path, no VGPR data) -----------
// GVS mode: 64-bit base in SGPRs, per-lane 32-bit byte offset in a VGPR,
// per-lane LDS byte offset in the VDST VGPR.
__device__ __forceinline__ void async_copy_b128(uint32_t lds_off,
                                                const void* gbase,
                                                uint32_t g_off) {
    asm volatile("global_load_async_to_lds_b128 %0, %1, %2"
                 :: "v"(lds_off), "v"(g_off), "s"(gbase)
                 : "memory");
}
__device__ __forceinline__ void wait_async0() {
    asm volatile("s_wait_asynccnt 0x0" ::: "memory");
}

// ---------------------------------------------------------------------------
// Kernel A: encoder MLPs (z0, params) + softplus + RK4 Lotka-Volterra scan.
// One thread per batch row. Weights staged transposed in LDS (broadcast reads).
// Two passes reuse one 55 KB LDS buffer (z-weights, then p-weights).
// ---------------------------------------------------------------------------
__global__ void __launch_bounds__(256)
encoder_rk4_kernel(const float* __restrict__ latent,
                   const float* __restrict__ latentp,
                   const float* __restrict__ tg,
                   const float* __restrict__ w1z, const float* __restrict__ b1z,
                   const float* __restrict__ w2z, const float* __restrict__ b2z,
                   const float* __restrict__ w1p, const float* __restrict__ b1p,
                   const float* __restrict__ w2p, const float* __restrict__ b2p,
                   float* __restrict__ out_z,   // (B, T, 2)
                   float* __restrict__ out_a, float* __restrict__ out_b,
                   float* __restrict__ out_c, float* __restrict__ out_d)
{
    extern __shared__ float smem[];
    float* w1T  = smem;               // HID*LAT, transposed: [j*LAT + k]
    float* b1s  = w1T + HID * LAT;    // HID
    float* w2s  = b1s + HID;          // HID*4  ([j*4 + n], zero padded)
    float* ts_s = w2s + HID * 4;      // TSTEP
    float* b2s  = ts_s + TSTEP;       // 4

    const int tid = threadIdx.x;
    const int r   = blockIdx.x * 256 + tid;

    // stage the time vector once
    for (int idx = tid; idx < TSTEP; idx += 256) ts_s[idx] = tg[idx];

    float resz[2] = {0.f, 0.f};
    float resp[4] = {0.f, 0.f, 0.f, 0.f};

    for (int pass = 0; pass < 2; ++pass) {
        const float* w1 = pass ? w1p : w1z;
        const float* b1 = pass ? b1p : b1z;
        const float* w2 = pass ? w2p : w2z;
        const float* b2 = pass ? b2p : b2z;
        const int    nw = pass ? 4 : 2;

        __syncthreads();   // previous pass done reading LDS
        for (int idx = tid; idx < HID * LAT; idx += 256) {
            int k = idx / HID, j = idx % HID;        // w1 is (LAT, HID)
            w1T[j * LAT + k] = w1[idx];
        }
        for (int idx = tid; idx < HID; idx += 256) {
            b1s[idx] = b1[idx];
            #pragma unroll
            for (int n = 0; n < 4; ++n)
                w2s[idx * 4 + n] = (n < nw) ? w2[idx * nw + n] : 0.0f;
        }
        if (tid < 4) b2s[tid] = (tid < nw) ? b2[tid] : 0.0f;
        __syncthreads();

        // load this row's 64-dim input into registers (16 x float4)
        float x[LAT];
        {
            const float4* xin = (const float4*)((pass ? latentp : latent) +
                                                (size_t)r * LAT);
            #pragma unroll
            for (int q = 0; q < LAT / 4; ++q) {
                float4 v = xin[q];
                x[4*q+0] = v.x; x[4*q+1] = v.y; x[4*q+2] = v.z; x[4*q+3] = v.w;
            }
        }

        float acc0 = b2s[0], acc1 = b2s[1], acc2 = b2s[2], acc3 = b2s[3];
        for (int j = 0; j < HID; ++j) {
            const float4* w4 = (const float4*)(w1T + j * LAT);
            float hj = b1s[j];
            #pragma unroll
            for (int q = 0; q < LAT / 4; ++q) {
                float4 w = w4[q];
                hj = fmaf(x[4*q+0], w.x, hj);
                hj = fmaf(x[4*q+1], w.y, hj);
                hj = fmaf(x[4*q+2], w.z, hj);
                hj = fmaf(x[4*q+3], w.w, hj);
            }
            hj = fmaxf(hj, 0.0f);                    // relu
            acc0 = fmaf(hj, w2s[j*4+0], acc0);
            acc1 = fmaf(hj, w2s[j*4+1], acc1);
            acc2 = fmaf(hj, w2s[j*4+2], acc2);
            acc3 = fmaf(hj, w2s[j*4+3], acc3);
        }
        if (pass == 0) { resz[0] = acc0; resz[1] = acc1; }
        else { resp[0] = acc0; resp[1] = acc1; resp[2] = acc2; resp[3] = acc3; }
    }

    float zx = softplus_f(resz[0]);
    float zy = softplus_f(resz[1]);
    float a  = softplus_f(resp[0]);
    float b  = softplus_f(resp[1]);
    float c  = softplus_f(resp[2]);
    float d  = softplus_f(resp[3]);

    out_a[r] = a; out_b[r] = b; out_c[r] = c; out_d[r] = d;

    // ---- RK4 scan over T-1 steps ----
    float* zr = out_z + (size_t)r * (TSTEP * 2);
    zr[0] = zx; zr[1] = zy;
    for (int i = 0; i < TSTEP - 1; ++i) {
        float dt  = ts_s[i + 1] - ts_s[i];
        float k1x = a*zx - b*zx*zy,       k1y = d*zx*zy - c*zy;
        float x2  = zx + 0.5f*dt*k1x,     y2  = zy + 0.5f*dt*k1y;
        float k2x = a*x2 - b*x2*y2,       k2y = d*x2*y2 - c*y2;
        float x3  = zx + 0.5f*dt*k2x,     y3  = zy + 0.5f*dt*k2y;
        float k3x = a*x3 - b*x3*y3,       k3y = d*x3*y3 - c*y3;
        float x4  = zx + dt*k3x,          y4  = zy + dt*k3y;
        float k4x = a*x4 - b*x4*y4,       k4y = d*x4*y4 - c*y4;
        zx += dt * (1.0f/6.0f) * (k1x + 2.f*k2x + 2.f*k3x + k4x);
        zy += dt * (1.0f/6.0f) * (k1y + 2.f*k2y + 2.f*k3y + k4y);
        zr[(i+1)*2 + 0] = zx;
        zr[(i+1)*2 + 1] = zy;
    }
}

// ---------------------------------------------------------------------------
// Kernel B: decoder MLP  (B*T, 2) -> relu(2->200) -> (200->128)
// Layer 2 runs on V_WMMA_F32_16X16X4_F32.
//   wave tile  : 16 rows x 128 cols, K = 200 in chunks of 4
//   block      : 8 waves = 128 rows;  grid = 819200/128 = 6400 blocks
// A fragment (the relu'd hidden acts) is computed in-register per lane
// directly in WMMA A layout (lane%16 = M row, VGPR j <-> K = 2*(lane/16)+j),
// so no barrier is needed inside the K loop.
// w2g is staged with GLOBAL_LOAD_ASYNC_TO_LDS_B128 (ASYNCcnt, no VGPR data
// path) in 2 K-phases of 100 rows; LDS row stride 136 floats so the two
// half-waves (rows K and K+2) hit disjoint bank groups.
// ---------------------------------------------------------------------------
constexpr int W2S_STRIDE = 136;   // floats; 544 bytes per staged row
constexpr int KPHASE     = 100;   // K rows per LDS staging phase

__global__ void __launch_bounds__(256)
decoder_wmma_kernel(const float* __restrict__ zall,   // (B*T, 2)
                    const float* __restrict__ w1g,    // (2, HID)
                    const float* __restrict__ b1g,    // (HID)
                    const float* __restrict__ w2g,    // (HID, INX)
                    const float* __restrict__ b2g,    // (INX)
                    float* __restrict__ out_x)        // (B*T, INX)
{
    extern __shared__ float smem[];
    float* w2s  = smem;                         // KPHASE * W2S_STRIDE
    float* w1gs = w2s + KPHASE * W2S_STRIDE;    // 2*HID
    float* b1gs = w1gs + 2 * HID;               // HID
    float* b2gs = b1gs + HID;                   // INX

    const int tid = threadIdx.x;
    for (int idx = tid; idx < 2 * HID; idx += 256) w1gs[idx] = w1g[idx];
    for (int idx = tid; idx < HID;     idx += 256) b1gs[idx] = b1g[idx];
    for (int idx = tid; idx < INX;     idx += 256) b2gs[idx] = b2g[idx];

    // LDS byte offset of w2s (low 32 bits of a generic shared pointer are the
    // LDS offset on CDNA5: aperture lives in addr[63:32]).
    const uint32_t w2sOff = (uint32_t)(size_t)(void*)w2s;

    const int lane = tid & 31;
    const int wave = tid >> 5;
    const int hw   = lane >> 4;      // half-wave: 0 or 1
    const int l16  = lane & 15;
    const int rowBase = blockIdx.x * 128 + wave * 16;
    const int row     = rowBase + l16;

    const float zx = zall[(size_t)row * 2 + 0];
    const float zy = zall[(size_t)row * 2 + 1];

    v8f acc[8];
    #pragma unroll
    for (int nt = 0; nt < 8; ++nt)
        #pragma unroll
        for (int e = 0; e < 8; ++e) acc[nt][e] = 0.0f;

    for (int p = 0; p < 2; ++p) {
        __syncthreads();   // protect w2s across phases

        // ---- async-stage 100 rows x 512B of w2g, keeping the 544B LDS
        // row stride (per-lane independent LDS destinations).
        for (int idx = tid; idx < KPHASE * (INX / 4); idx += 256) {
            uint32_t lds = w2sOff + (uint32_t)((idx >> 5) * (W2S_STRIDE * 4)
                                               + (idx & 31) * 16);
            uint32_t gof = (uint32_t)(p * KPHASE * INX * 4 + idx * 16);
            async_copy_b128(lds, (const void*)w2g, gof);
        }
        wait_async0();
        __syncthreads();

        for (int k = 0; k < KPHASE; k += 4) {
            const int kg = p * KPHASE + k;     // global K base of this chunk
            // ---- A fragment: h = relu(zx*w1g[0,c] + zy*w1g[1,c] + b1g[c])
            v2f afrag;
            #pragma unroll
            for (int j = 0; j < 2; ++j) {
                int cc = kg + hw * 2 + j;
                float h = fmaf(zx, w1gs[cc], fmaf(zy, w1gs[HID + cc], b1gs[cc]));
                afrag[j] = fmaxf(h, 0.0f);
            }
            // ---- 8 column tiles of 16
            #pragma unroll
            for (int nt = 0; nt < 8; ++nt) {
                const int col = nt * 16 + l16;
                v2f bfrag;
                bfrag[0] = w2s[(k + hw * 2 + 0) * W2S_STRIDE + col];
                bfrag[1] = w2s[(k + hw * 2 + 1) * W2S_STRIDE + col];
                acc[nt] = __builtin_amdgcn_wmma_f32_16x16x4_f32(
                    false, afrag, false, bfrag, (short)0, acc[nt], false, false);
            }
        }
    }

    // ---- epilogue: + b2g, store. C layout: VGPR v, lanes 0-15 -> M=v,
    // lanes 16-31 -> M=v+8, N = lane%16.
    #pragma unroll
    for (int nt = 0; nt < 8; ++nt) {
        const int col = nt * 16 + l16;
        const float bb = b2gs[col];
        float* orow = out_x + (size_t)(rowBase + 8 * hw) * INX + col;
        #pragma unroll
        for (int v = 0; v < 8; ++v)
            orow[(size_t)v * INX] = acc[nt][v] + bb;
    }
}

// ---------------------------------------------------------------------------
extern "C" void kernel_launch(void* const* d_in, const int* in_sizes, int n_in,
                              void* d_out, int out_size, void* d_ws, size_t ws_size,
                              hipStream_t stream) {
    (void)in_sizes; (void)n_in; (void)out_size; (void)d_ws; (void)ws_size;

    const float* latent  = (const float*)d_in[0];
    const float* latentp = (const float*)d_in[1];
    const float* tg      = (const float*)d_in[2];
    const float* w1z = (const float*)d_in[3];
    const float* b1z = (const float*)d_in[4];
    const float* w2z = (const float*)d_in[5];
    const float* b2z = (const float*)d_in[6];
    const float* w1p = (const float*)d_in[7];
    const float* b1p = (const float*)d_in[8];
    const float* w2p = (const float*)d_in[9];
    const float* b2p = (const float*)d_in[10];
    const float* w1g = (const float*)d_in[11];
    const float* b1g = (const float*)d_in[12];
    const float* w2g = (const float*)d_in[13];
    const float* b2g = (const float*)d_in[14];

    float* out   = (float*)d_out;
    float* out_x = out;                                       // B*T*128
    float* out_z = out_x + (size_t)BATCH * TSTEP * INX;       // B*T*2
    float* out_a = out_z + (size_t)BATCH * TSTEP * 2;
    float* out_b = out_a + BATCH;
    float* out_c = out_b + BATCH;
    float* out_d = out_c + BATCH;

    const size_t smemA = (size_t)(HID * LAT + HID + HID * 4 + TSTEP + 4)
                         * sizeof(float);
    encoder_rk4_kernel<<<BATCH / 256, 256, smemA, stream>>>(
        latent, latentp, tg,
        w1z, b1z, w2z, b2z,
        w1p, b1p, w2p, b2p,
        out_z, out_a, out_b, out_c, out_d);

    const size_t smemB = (size_t)(KPHASE * W2S_STRIDE + 2 * HID + HID + INX)
                         * sizeof(float);
    decoder_wmma_kernel<<<(BATCH * TSTEP) / 128, 256, smemB, stream>>>(
        out_z, w1g, b1g, w2g, b2g, out_x);
}